// LRULayer_55843164782905
// MI455X (gfx1250) — compile-verified
//
#include <hip/hip_runtime.h>
#include <cstdint>

// ---------------- problem constants ----------------
#define Hdim 256   // D_MODEL
#define NST  64    // D_STATE
#define Bb   4
#define Lseq 8192
#define Tc   128   // chunk length
#define NCH  (Lseq / Tc)   // 64 chunks
#define HHALF 128  // channel-half tile width for LDS staging

typedef __attribute__((ext_vector_type(2))) float v2f;
typedef __attribute__((ext_vector_type(8))) float v8f;
typedef __attribute__((ext_vector_type(4))) unsigned int v4u;
typedef __attribute__((ext_vector_type(4))) int v4i;
typedef __attribute__((ext_vector_type(8))) int v8i;

// ---- WMMA f32 16x16x4 (gfx1250) ----
__device__ __forceinline__ v8f wmma4_f32(v2f a, v2f b, v8f c) {
  return __builtin_amdgcn_wmma_f32_16x16x4_f32(false, a, false, b, (short)0, c,
                                               false, false);
}

// A fragment: A stored packed as float4 blocks af[k4][row] = A[row][4k..4k+3].
__device__ __forceinline__ v2f ld_afrag(const float* __restrict__ apack, int k4idx,
                                        int mrows, int mbase, int lane) {
  int m = mbase + (lane & 15);
  const float* p = apack + (((size_t)k4idx * mrows + m) << 2) + ((lane >> 4) << 1);
  v2f r; r[0] = p[0]; r[1] = p[1];
  return r;
}

// B fragment from a row-major (K x ldb) matrix (global or LDS).
__device__ __forceinline__ v2f ld_bfrag(const float* Bm, int ldb, int kbase, int col,
                                        int lane) {
  int k = kbase + ((lane >> 4) << 1);
  v2f r;
  r[0] = Bm[(size_t)k * ldb + col];
  r[1] = Bm[(size_t)(k + 1) * ldb + col];
  return r;
}

// ---- Tensor Data Mover: 2D strided tile (global -> LDS), fp32 elements ----
// Loads `rows` rows of `row_elems` floats, global row stride `row_stride_elems`,
// packed contiguously into LDS at byte offset lds_off. Tracked by TENSORcnt.
__device__ __forceinline__ void tdm_load_2d(unsigned lds_off, const void* gsrc,
                                            unsigned row_elems, unsigned rows,
                                            unsigned row_stride_elems) {
  unsigned long long ga = (unsigned long long)(uintptr_t)gsrc;
  v4u g0;
  g0[0] = 1u;                                   // count=1 (valid user descriptor)
  g0[1] = lds_off;                              // lds_addr (bytes)
  g0[2] = (unsigned)ga;                         // global_addr[31:0]
  g0[3] = (unsigned)((ga >> 32) & 0x01ffffffu)  // global_addr[56:32]
          | (2u << 30);                         // type = 2 ("image")
  v8i g1;
  g1[0] = (int)(2u << 16);                      // data_size = 2 -> 4 bytes; mask=0
  g1[1] = (int)((row_stride_elems & 0xffffu) << 16);            // tensor_dim0 lo16
  g1[2] = (int)(((row_stride_elems >> 16) & 0xffffu)            // tensor_dim0 hi16
                | ((rows & 0xffffu) << 16));                    // tensor_dim1 lo16
  g1[3] = (int)(((rows >> 16) & 0xffffu)                        // tensor_dim1 hi16
                | ((row_elems & 0xffffu) << 16));               // tile_dim0
  g1[4] = (int)(rows & 0xffffu);                                // tile_dim1; tile_dim2=0
  g1[5] = (int)row_stride_elems;                                // tensor_dim0_stride lo32
  g1[6] = 0;                                                    // stride hi / dim1_stride
  g1[7] = 0;
  v4i gz = {0, 0, 0, 0};
#if __clang_major__ >= 23
  v8i gz8 = {0, 0, 0, 0, 0, 0, 0, 0};
  __builtin_amdgcn_tensor_load_to_lds(g0, g1, gz, gz, gz8, 0);
#else
  __builtin_amdgcn_tensor_load_to_lds(g0, g1, gz, gz, 0);
#endif
}

// ---------------- precompute 1: lambda powers, packed A matrices ----------------
__global__ void lru_precompute1(const float* __restrict__ nu_log,
                                const float* __restrict__ theta_log,
                                float* __restrict__ lamT_re, float* __restrict__ lamT_im,
                                float* __restrict__ powre, float* __restrict__ powim,
                                float* __restrict__ P2re_pk, float* __restrict__ P2im_pk,
                                float* __restrict__ Acar_pk) {
  int n = threadIdx.x;
  if (n >= NST) return;
  float nu = __expf(nu_log[n]);
  float th = __expf(theta_log[n]);
  float rad = __expf(-nu);
  float lre = rad * __cosf(th);
  float lim = rad * __sinf(th);
  float pre = 1.f, pim = 0.f;  // lambda^0
  for (int j = 0; j <= Tc; ++j) {
    powre[(size_t)j * NST + n] = pre;
    powim[(size_t)j * NST + n] = pim;
    if (j <= Tc - 1) {
      int k = Tc - 1 - j;  // P2[n][k] = lambda^(Tc-1-k)
      size_t pidx = (((size_t)(k >> 2) * NST + n) << 2) + (k & 3);
      P2re_pk[pidx] = pre;
      P2im_pk[pidx] = pim;
    }
    if (j >= 1) {
      int jr = j - 1;  // Acar row jr uses lambda^(jr+1)
      int kc0 = n, kc1 = NST + n;
      Acar_pk[(((size_t)(kc0 >> 2) * Tc + jr) << 2) + (kc0 & 3)] = pre;
      Acar_pk[(((size_t)(kc1 >> 2) * Tc + jr) << 2) + (kc1 & 3)] = -pim;
    }
    if (j == Tc) { lamT_re[n] = pre; lamT_im[n] = pim; }
    float t = pre * lre - pim * lim;
    pim = pre * lim + pim * lre;
    pre = t;
  }
}

// ---------------- precompute 2: CB (transposed) and local kernel K[l][h] -------------
__global__ void lru_precompute2(const float* __restrict__ B_re, const float* __restrict__ B_im,
                                const float* __restrict__ C_re, const float* __restrict__ C_im,
                                const float* __restrict__ powre, const float* __restrict__ powim,
                                float* __restrict__ CBt_re, float* __restrict__ CBt_im,
                                float* __restrict__ Kloc) {
  int tid = blockIdx.x * blockDim.x + threadIdx.x;  // 0 .. Tc*Hdim-1
  int l = tid / Hdim, h = tid % Hdim;
  float acc = 0.f;
  for (int n = 0; n < NST; ++n) {
    float br = B_re[(size_t)h * NST + n], bi = B_im[(size_t)h * NST + n];
    float cr = C_re[(size_t)h * NST + n], ci = C_im[(size_t)h * NST + n];
    float cbre = cr * br - ci * bi;
    float cbim = cr * bi + ci * br;
    acc += cbre * powre[(size_t)l * NST + n] - cbim * powim[(size_t)l * NST + n];
  }
  Kloc[(size_t)l * Hdim + h] = acc;
  if (l < NST) {
    int n = l;
    float br = B_re[(size_t)h * NST + n], bi = B_im[(size_t)h * NST + n];
    float cr = C_re[(size_t)h * NST + n], ci = C_im[(size_t)h * NST + n];
    CBt_re[(size_t)n * Hdim + h] = cr * br - ci * bi;
    CBt_im[(size_t)n * Hdim + h] = cr * bi + ci * br;
  }
}

// ---------------- pass 1: chunk summaries U = P2 @ X  (TDM stage + WMMA) -------------
// Block = (b, chunk, channel-half). X tile (Tc x 128) staged via TDM into LDS.
__global__ void lru_pass1(const float* __restrict__ x,
                          const float* __restrict__ P2re_pk, const float* __restrict__ P2im_pk,
                          float* __restrict__ Ure, float* __restrict__ Uim) {
  __shared__ float Xs[Tc * HHALF];  // 64 KB, row-major [Tc][128]
  int bid = blockIdx.x;             // 0 .. Bb*NCH*2-1
  int bm = bid >> 1;
  int h0 = (bid & 1) * HHALF;
  int b = bm / NCH, m = bm % NCH;
  int wave = threadIdx.x >> 5, lane = threadIdx.x & 31;
  const float* Xg = x + ((size_t)b * Lseq + (size_t)m * Tc) * Hdim + h0;
  if (wave == 0) {
    tdm_load_2d((unsigned)(uintptr_t)Xs, Xg, HHALF, Tc, Hdim);
    __builtin_amdgcn_s_wait_tensorcnt(0);
  }
  __syncthreads();
  for (int tile = wave; tile < 64; tile += 8) {
    int part = tile >> 5;        // 0: real, 1: imag
    int rt = (tile >> 3) & 3;    // n-tile (4)
    int ct = tile & 7;           // h-tile (8)
    const float* ap = part ? P2im_pk : P2re_pk;
    float* up = part ? Uim : Ure;
    int colL = ct * 16 + (lane & 15);
    v8f acc = {0.f, 0.f, 0.f, 0.f, 0.f, 0.f, 0.f, 0.f};
#pragma unroll 8
    for (int k4 = 0; k4 < Tc / 4; ++k4) {
      v2f a = ld_afrag(ap, k4, NST, rt * 16, lane);
      v2f bf = ld_bfrag(Xs, HHALF, k4 * 4, colL, lane);
      acc = wmma4_f32(a, bf, acc);
    }
    int nb = rt * 16 + ((lane >> 4) << 3);
    float* dst = up + (size_t)bm * NST * Hdim + h0;
#pragma unroll
    for (int r = 0; r < 8; ++r) dst[(size_t)(nb + r) * Hdim + colL] = acc[r];
  }
}

// ---------------- pass 2: serial carry scan; rewrite U in place as W = CB * c_in -----
__global__ void lru_pass2(const float* __restrict__ lamT_re, const float* __restrict__ lamT_im,
                          const float* __restrict__ CBt_re, const float* __restrict__ CBt_im,
                          float* __restrict__ Ure, float* __restrict__ Uim) {
  int tid = blockIdx.x * blockDim.x + threadIdx.x;  // 0 .. NST*Hdim-1
  int n = tid / Hdim, h = tid % Hdim;
  float ltr = lamT_re[n], lti = lamT_im[n];
  float cbre = CBt_re[(size_t)n * Hdim + h];
  float cbim = CBt_im[(size_t)n * Hdim + h];
  for (int b = 0; b < Bb; ++b) {
    float cre = 0.f, cim = 0.f;
    for (int m = 0; m < NCH; ++m) {
      size_t idx = (((size_t)(b * NCH + m)) * NST + n) * Hdim + h;
      float ur = Ure[idx], ui = Uim[idx];
      Ure[idx] = cbre * cre - cbim * cim;  // W_re for chunk m
      Uim[idx] = cbre * cim + cbim * cre;  // W_im for chunk m
      float t = ltr * cre - lti * cim + ur;
      cim = ltr * cim + lti * cre + ui;
      cre = t;
    }
  }
}

// ---------------- pass 3a: y_carry = [P1re | -P1im] @ [Wre; Wim]  (TDM + WMMA) -------
// W (128 x 128 half-tile) staged compactly in LDS by two TDM descriptors.
__global__ void lru_pass3a(const float* __restrict__ Acar_pk,
                           const float* __restrict__ Ure, const float* __restrict__ Uim,
                           float* __restrict__ out) {
  __shared__ float Ws[2 * NST * HHALF];  // 64 KB: rows 0..63 = Wre, 64..127 = Wim
  int bid = blockIdx.x;
  int bm = bid >> 1;
  int h0 = (bid & 1) * HHALF;
  int b = bm / NCH, m = bm % NCH;
  int wave = threadIdx.x >> 5, lane = threadIdx.x & 31;
  if (wave == 0) {
    unsigned base = (unsigned)(uintptr_t)Ws;
    tdm_load_2d(base, Ure + (size_t)bm * NST * Hdim + h0, HHALF, NST, Hdim);
    tdm_load_2d(base + NST * HHALF * 4, Uim + (size_t)bm * NST * Hdim + h0, HHALF, NST,
                Hdim);
    __builtin_amdgcn_s_wait_tensorcnt(0);
  }
  __syncthreads();
  for (int tile = wave; tile < 64; tile += 8) {
    int rt = tile >> 3;   // j-tile (8)
    int ct = tile & 7;    // h-tile (8)
    int colL = ct * 16 + (lane & 15);
    v8f acc = {0.f, 0.f, 0.f, 0.f, 0.f, 0.f, 0.f, 0.f};
#pragma unroll 8
    for (int k4 = 0; k4 < 32; ++k4) {  // K = 2*NST = 128, compact in LDS
      v2f a = ld_afrag(Acar_pk, k4, Tc, rt * 16, lane);
      v2f bf = ld_bfrag(Ws, HHALF, k4 * 4, colL, lane);
      acc = wmma4_f32(a, bf, acc);
    }
    int jb = rt * 16 + ((lane >> 4) << 3);
    float* dst = out + ((size_t)b * Lseq + (size_t)m * Tc) * Hdim + h0;
#pragma unroll
    for (int r = 0; r < 8; ++r) dst[(size_t)(jb + r) * Hdim + colL] = acc[r];
  }
}

// ---------------- pass 3b: intra-chunk triangular Toeplitz + D*x epilogue ------------
__global__ void lru_pass3b(const float* __restrict__ x, const float* __restrict__ Kloc,
                           const float* __restrict__ Dv, float* __restrict__ out) {
  int bm = blockIdx.x;
  int b = bm / NCH, m = bm % NCH;
  int h = threadIdx.x;
  const float* Xc = x + ((size_t)b * Lseq + (size_t)m * Tc) * Hdim;
  float* Yc = out + ((size_t)b * Lseq + (size_t)m * Tc) * Hdim;
  float dv = Dv[h];
  for (int jt = 0; jt < Tc / 16; ++jt) {
    int j0 = jt * 16;
    float acc[16];
#pragma unroll
    for (int jj = 0; jj < 16; ++jj) acc[jj] = 0.f;
    for (int k = 0; k < j0; ++k) {
      float xk = Xc[(size_t)k * Hdim + h];
#pragma unroll
      for (int jj = 0; jj < 16; ++jj)
        acc[jj] += Kloc[(size_t)(j0 + jj - k) * Hdim + h] * xk;
    }
    for (int kk = 0; kk < 16; ++kk) {
      int k = j0 + kk;
      float xk = Xc[(size_t)k * Hdim + h];
#pragma unroll
      for (int jj = 0; jj < 16; ++jj)
        if (jj >= kk) acc[jj] += Kloc[(size_t)(jj - kk) * Hdim + h] * xk;
    }
#pragma unroll
    for (int jj = 0; jj < 16; ++jj) {
      int j = j0 + jj;
      size_t oi = (size_t)j * Hdim + h;
      Yc[oi] = Yc[oi] + acc[jj] + dv * Xc[oi];
    }
  }
}

// ---------------- launch ----------------
extern "C" void kernel_launch(void* const* d_in, const int* in_sizes, int n_in,
                              void* d_out, int out_size, void* d_ws, size_t ws_size,
                              hipStream_t stream) {
  (void)in_sizes; (void)n_in; (void)out_size; (void)ws_size;
  const float* x         = (const float*)d_in[0];
  const float* nu_log    = (const float*)d_in[1];
  const float* theta_log = (const float*)d_in[2];
  const float* B_re      = (const float*)d_in[3];
  const float* B_im      = (const float*)d_in[4];
  const float* C_re      = (const float*)d_in[5];
  const float* C_im      = (const float*)d_in[6];
  const float* Dv        = (const float*)d_in[7];
  float* out = (float*)d_out;

  // workspace layout (floats)
  float* ws = (float*)d_ws;
  float* lamT_re = ws;
  float* lamT_im = lamT_re + NST;
  float* powre   = lamT_im + NST;
  float* powim   = powre + (size_t)(Tc + 1) * NST;
  float* P2re_pk = powim + (size_t)(Tc + 1) * NST;
  float* P2im_pk = P2re_pk + (size_t)Tc * NST;
  float* Acar_pk = P2im_pk + (size_t)Tc * NST;
  float* CBt_re  = Acar_pk + (size_t)2 * NST * Tc;
  float* CBt_im  = CBt_re + (size_t)NST * Hdim;
  float* Kloc    = CBt_im + (size_t)NST * Hdim;
  float* Ure     = Kloc + (size_t)Tc * Hdim;
  float* Uim     = Ure + (size_t)Bb * NCH * NST * Hdim;

  lru_precompute1<<<1, 64, 0, stream>>>(nu_log, theta_log, lamT_re, lamT_im,
                                        powre, powim, P2re_pk, P2im_pk, Acar_pk);
  lru_precompute2<<<(Tc * Hdim) / 256, 256, 0, stream>>>(B_re, B_im, C_re, C_im,
                                                         powre, powim,
                                                         CBt_re, CBt_im, Kloc);
  lru_pass1<<<Bb * NCH * 2, 256, 0, stream>>>(x, P2re_pk, P2im_pk, Ure, Uim);
  lru_pass2<<<(NST * Hdim) / 256, 256, 0, stream>>>(lamT_re, lamT_im, CBt_re, CBt_im,
                                                    Ure, Uim);
  lru_pass3a<<<Bb * NCH * 2, 256, 0, stream>>>(Acar_pk, Ure, Uim, out);
  lru_pass3b<<<Bb * NCH, 256, 0, stream>>>(x, Kloc, Dv, out);
}